// SparseGCNLayer_65635690218019
// MI455X (gfx1250) — compile-verified
//
#include <hip/hip_runtime.h>

// ---------------------------------------------------------------------------
// SparseGCN layer for MI455X (gfx1250, wave32):
//   h = scatter_add(edge_val * x[edge_src] -> edge_dst)   [memory/atomic bound]
//   out = relu(h @ W.T + b)                               [V_WMMA_F32_16X16X4_F32]
// ---------------------------------------------------------------------------

#define N_NODES 50000
#define N_EDGES 800000
#define D_FEAT  128

typedef float v2f __attribute__((ext_vector_type(2)));
typedef float v8f __attribute__((ext_vector_type(8)));

// --- Kernel 1: zero the h accumulator (d_ws is poisoned, graph replays need
// --- deterministic re-zero every call). 1.6M float4 stores.
__global__ void gcn_zero_h(float4* __restrict__ h4) {
    int i = blockIdx.x * blockDim.x + threadIdx.x;
    h4[i] = make_float4(0.f, 0.f, 0.f, 0.f);
}

// --- Kernel 2: edge scatter. One wave32 per edge; each lane owns 4 contiguous
// --- features (32*4 = 128). x-row gather is a coalesced float4 load (L2-hot,
// --- x is 25.6MB << 192MB L2); accumulation via f32 global atomics that
// --- resolve in L2 (h is L2-resident too).
__global__ void gcn_scatter(const float* __restrict__ x,
                            const int*   __restrict__ esrc,
                            const int*   __restrict__ edst,
                            const float* __restrict__ eval,
                            float*       __restrict__ h) {
    int gid  = blockIdx.x * blockDim.x + threadIdx.x;
    int e    = gid >> 5;          // edge index
    int lane = gid & 31;
    if (e >= N_EDGES) return;
    int   src = esrc[e];
    int   dst = edst[e];
    float val = eval[e];
    float4 v = *(const float4*)(x + src * D_FEAT + lane * 4);
    float* hd = h + dst * D_FEAT + lane * 4;
    atomicAdd(hd + 0, val * v.x);
    atomicAdd(hd + 1, val * v.y);
    atomicAdd(hd + 2, val * v.z);
    atomicAdd(hd + 3, val * v.w);
}

// --- Kernel 3: out = relu(h @ W.T + b) via V_WMMA_F32_16X16X4_F32.
// --- Block = 256 threads = 8 waves; wave w computes the 16x16 tile at
// --- (rows blockIdx.x*16, cols w*16). K loop: 128/4 = 32 WMMA issues.
// ---
// --- A (16x4 f32) lane layout: row = lane&15, K = k0 + 2*(lane>>4) + {0,1}
// --- B (4x16 f32) lane layout: col = lane&15, K = k0 + 2*(lane>>4) + {0,1}
// ---   B[k][n] = W[n][k] -> consecutive k pair = float2 from W row n.
// --- C/D (16x16 f32): VGPR v -> row v + 8*(lane>>4), col = lane&15.
__global__ void __launch_bounds__(256)
gcn_gemm_bias_relu(const float* __restrict__ h,
                   const float* __restrict__ W,
                   const float* __restrict__ bias,
                   float*       __restrict__ out) {
    const int wave = threadIdx.x >> 5;     // 0..7 -> column tile
    const int lane = threadIdx.x & 31;
    const int row0 = blockIdx.x * 16;      // node tile base (exact: 50000/16)
    const int n0   = wave * 16;            // output-feature tile base
    const int m    = lane & 15;            // row (A) / col (B) within tile
    const int hi   = lane >> 4;            // half-wave selector
    const int koff = hi * 2;

    const float* arow = h + (row0 + m) * D_FEAT;  // A fragment source row
    const float* brow = W + (n0   + m) * D_FEAT;  // B fragment source row (W.T col)

    v8f c = {};
#pragma unroll
    for (int k0 = 0; k0 < D_FEAT; k0 += 4) {
        v2f a = *(const v2f*)(arow + k0 + koff);
        v2f b = *(const v2f*)(brow + k0 + koff);
        // (neg_a, A, neg_b, B, c_mod, C, reuse_a, reuse_b)
        c = __builtin_amdgcn_wmma_f32_16x16x4_f32(false, a, false, b,
                                                  (short)0, c, false, false);
    }

    const float bv = bias[n0 + m];         // bias indexed by output column
#pragma unroll
    for (int v = 0; v < 8; ++v) {
        float r = c[v] + bv;
        r = fmaxf(r, 0.0f);                // ReLU
        out[(row0 + v + 8 * hi) * D_FEAT + n0 + m] = r;
    }
}

extern "C" void kernel_launch(void* const* d_in, const int* in_sizes, int n_in,
                              void* d_out, int out_size, void* d_ws, size_t ws_size,
                              hipStream_t stream) {
    const float* x    = (const float*)d_in[0];
    const int*   esrc = (const int*)  d_in[1];
    const int*   edst = (const int*)  d_in[2];
    const float* eval = (const float*)d_in[3];
    const float* W    = (const float*)d_in[4];
    const float* b    = (const float*)d_in[5];
    float*       out  = (float*)d_out;
    float*       h    = (float*)d_ws;   // N_NODES * D_FEAT * 4B = 25.6 MB

    // 1) zero accumulator: 50000*128/4 = 1,600,000 float4 -> 6250 blocks exact
    gcn_zero_h<<<(N_NODES * D_FEAT / 4) / 256, 256, 0, stream>>>((float4*)h);

    // 2) scatter: 800000 edges * 32 lanes = 25.6M threads -> 100000 blocks exact
    gcn_scatter<<<(N_EDGES * 32) / 256, 256, 0, stream>>>(x, esrc, edst, eval, h);

    // 3) dense layer: 3125 row tiles, 8 waves/block cover all 8 col tiles
    gcn_gemm_bias_relu<<<N_NODES / 16, 256, 0, stream>>>(h, W, b, out);
}